// G_90486370992839
// MI455X (gfx1250) — compile-verified
//
#include <hip/hip_runtime.h>

// ---------------------------------------------------------------------------
// Char-LSTM forward for MI455X (gfx1250, wave32, WMMA).
//
//   emb+input-proj collapse to a 26x104 lookup table XP (since x in [0,26)).
//   Per step:   gates[16,104] = h[16,26] @ W_hh^T  (V_WMMA_F32_16X16X4_F32)
//               + XP[x[b,t]]  then sigmoid/tanh cell update.
//   1 workgroup (128 thr = 4 waves) per 16-batch tile; waves split the 8
//   (padded) N-tiles; barriers x2 per step; c in regs, h/gates in LDS.
// ---------------------------------------------------------------------------

typedef __attribute__((ext_vector_type(2))) float v2f;
typedef __attribute__((ext_vector_type(4))) float v4f;
typedef __attribute__((ext_vector_type(8))) float v8f;

#define HH   26          // hidden size
#define GG   104         // 4*H gate columns
#define NTT  8           // padded N tiles (128 cols)
#define KTT  7           // K slices of 4 (28, h rows padded w/ zeros)
#define TSTEPS 1024
#define TCH  32          // x-index staging chunk (timesteps)

__device__ __forceinline__ float fsig(float x) {
    // sigmoid(x) = 1 / (1 + 2^(-x*log2(e)))   -> v_exp_f32 + v_rcp_f32
    return __builtin_amdgcn_rcpf(1.0f + __builtin_amdgcn_exp2f(-1.44269504088896f * x));
}
__device__ __forceinline__ float ftanh(float x) {
    // tanh(x) = 2*sigmoid(2x) - 1
    return __builtin_fmaf(2.0f, fsig(2.0f * x), -1.0f);
}

__global__ __launch_bounds__(128) void lstm_char_fwd(
    const int*  __restrict__ x,      // [B, T] token ids
    const float* __restrict__ E,     // [26, 26]
    const float* __restrict__ W_ih,  // [104, 26]
    const float* __restrict__ W_hh,  // [104, 26]
    const float* __restrict__ b_ih,  // [104]
    const float* __restrict__ b_hh,  // [104]
    const float* __restrict__ W_lin, // [26, 26]
    const float* __restrict__ b_lin, // [26]
    float* __restrict__ out,         // [B, 26]
    int T)
{
    __shared__ float xp_lds[26 * 26 * 4];        // XP table, [c][j][gate(i,f,g,o)]
    __shared__ float gates_lds[NTT * 16 * 16];   // [gate_col(128)][m(16)]
    __shared__ float h_lds[16 * 28];             // [m][k], k padded 26->28 w/ zeros
    __shared__ int   x_lds[16 * TCH];            // staged token ids

    const int tid   = threadIdx.x;
    const int lane  = tid & 31;
    const int wv    = tid >> 5;        // wave 0..3
    const int nloc  = lane & 15;       // WMMA: A row / B,D column within tile
    const int khalf = lane >> 4;       // WMMA: K/M half select
    const int tile  = blockIdx.x;      // 16-batch tile

    // ---- build XP[c][j][gate] = b_ih[g]+b_hh[g] + sum_h E[c,h]*W_ih[g,h] ----
    for (int idx = tid; idx < 26 * 26; idx += 128) {
        const int c = idx / 26, j = idx - c * 26;
        v4f v;
        #pragma unroll
        for (int gate = 0; gate < 4; ++gate) {
            const int g = gate * 26 + j;
            float s = b_ih[g] + b_hh[g];
            #pragma unroll
            for (int h = 0; h < 26; ++h)
                s = __builtin_fmaf(E[c * 26 + h], W_ih[g * 26 + h], s);
            v[gate] = s;
        }
        *(v4f*)&xp_lds[idx * 4] = v;
    }
    // zero h (incl. K padding rows 26,27 which must stay 0 forever)
    for (int i = tid; i < 16 * 28; i += 128) h_lds[i] = 0.0f;

    // ---- preload loop-invariant B fragments: B[k][n] = W_hh[n][k] ----------
    // B 4x16 layout (mirror of A): vgpr v holds K = khalf*2 + v, N = nloc.
    v2f bfrag[2][KTT];
    #pragma unroll
    for (int s = 0; s < 2; ++s) {
        const int n_abs = (wv + s * 4) * 16 + nloc;
        #pragma unroll
        for (int kt = 0; kt < KTT; ++kt) {
            const int k0 = kt * 4 + khalf * 2;
            v2f b = {0.0f, 0.0f};
            if (n_abs < GG) {
                if (k0     < HH) b[0] = W_hh[n_abs * HH + k0];
                if (k0 + 1 < HH) b[1] = W_hh[n_abs * HH + k0 + 1];
            }
            bfrag[s][kt] = b;
        }
    }

    // elementwise mapping: thread -> (batch row m, hidden set {g8, g8+8, ...})
    const int m  = tid & 15;
    const int g8 = tid >> 4;           // 0..7
    float creg[4] = {0.0f, 0.0f, 0.0f, 0.0f};

    __syncthreads();

    for (int t = 0; t < TSTEPS; ++t) {
        if ((t & (TCH - 1)) == 0) {
            // stage x[tile*16 + m][t .. t+31] cooperatively (coalesced b128)
            const int toff = g8 * 4;
            *(int4*)&x_lds[m * TCH + toff] =
                *(const int4*)&x[(tile * 16 + m) * T + t + toff];
            __syncthreads();
        }

        // ---- gates = h @ W_hh^T via f32 WMMA (this wave: N-tiles wv, wv+4) --
        v8f acc0 = {}; v8f acc1 = {};
        #pragma unroll
        for (int kt = 0; kt < KTT; ++kt) {
            // A 16x4 layout: lanes0-15 M=lane K={0,1}; lanes16-31 M=lane-16 K={2,3}
            const v2f a = *(const v2f*)&h_lds[nloc * 28 + kt * 4 + khalf * 2];
            acc0 = __builtin_amdgcn_wmma_f32_16x16x4_f32(
                       false, a, false, bfrag[0][kt], (short)0, acc0, false, false);
            acc1 = __builtin_amdgcn_wmma_f32_16x16x4_f32(
                       false, a, false, bfrag[1][kt], (short)0, acc1, false, false);
        }
        // D layout: lane holds col n=nloc, rows m = khalf*8 .. khalf*8+7
        {
            const int mb = khalf * 8;
            float* d0 = &gates_lds[((wv    ) * 16 + nloc) * 16 + mb];
            float* d1 = &gates_lds[((wv + 4) * 16 + nloc) * 16 + mb];
            *(v4f*)(d0)     = (v4f){acc0[0], acc0[1], acc0[2], acc0[3]};
            *(v4f*)(d0 + 4) = (v4f){acc0[4], acc0[5], acc0[6], acc0[7]};
            *(v4f*)(d1)     = (v4f){acc1[0], acc1[1], acc1[2], acc1[3]};
            *(v4f*)(d1 + 4) = (v4f){acc1[4], acc1[5], acc1[6], acc1[7]};
        }
        __syncthreads();

        // ---- LSTM cell update ----------------------------------------------
        const int ch = x_lds[m * TCH + (t & (TCH - 1))];
        #pragma unroll
        for (int r = 0; r < 4; ++r) {
            const int j = g8 + 8 * r;
            if (j < HH) {
                const v4f xp = *(const v4f*)&xp_lds[(ch * 26 + j) * 4];
                const float gi = fsig (gates_lds[(      j) * 16 + m] + xp[0]);
                const float gf = fsig (gates_lds[(26  + j) * 16 + m] + xp[1]);
                const float gc = ftanh(gates_lds[(52  + j) * 16 + m] + xp[2]);
                const float go = fsig (gates_lds[(78  + j) * 16 + m] + xp[3]);
                const float c  = __builtin_fmaf(gf, creg[r], gi * gc);
                creg[r] = c;
                h_lds[m * 28 + j] = go * ftanh(c);
            }
        }
        __syncthreads();   // h ready for next step's A fragments
    }

    // ---- final linear: out[b][n] = b_lin[n] + sum_j h[m][j] * W_lin[n][j] --
    #pragma unroll
    for (int r = 0; r < 4; ++r) {
        const int n = g8 + 8 * r;
        if (n < HH) {
            float s = b_lin[n];
            #pragma unroll
            for (int j = 0; j < 26; ++j)
                s = __builtin_fmaf(h_lds[m * 28 + j], W_lin[n * 26 + j], s);
            out[(tile * 16 + m) * 26 + n] = s;
        }
    }
}

extern "C" void kernel_launch(void* const* d_in, const int* in_sizes, int n_in,
                              void* d_out, int out_size, void* d_ws, size_t ws_size,
                              hipStream_t stream) {
    const int*   x     = (const int*)  d_in[0];
    const float* E     = (const float*)d_in[1];
    const float* W_ih  = (const float*)d_in[2];
    const float* W_hh  = (const float*)d_in[3];
    const float* b_ih  = (const float*)d_in[4];
    const float* b_hh  = (const float*)d_in[5];
    const float* W_lin = (const float*)d_in[6];
    const float* b_lin = (const float*)d_in[7];
    float* out = (float*)d_out;

    const int T = TSTEPS;
    const int B = in_sizes[0] / T;        // 2048
    const int tiles = B / 16;             // 128 workgroups, 4 waves each

    lstm_char_fwd<<<tiles, 128, 0, stream>>>(x, E, W_ih, W_hh, b_ih, b_hh,
                                             W_lin, b_lin, out, T);
    (void)d_ws; (void)ws_size; (void)n_in; (void)out_size;
}